// OneHot_26199300506262
// MI455X (gfx1250) — compile-verified
//
#include <hip/hip_runtime.h>
#include <hip/hip_bf16.h>
#include <stdint.h>

// Problem geometry (fixed by the reference):
//   x: (2,8,2048,2048) f32 -> 64 matrices of 1024x1024, H=W=32 grid.
#define N_MAT      64
#define HW_DIM     1024
#define ROW_BYTES  8192u          // 2048 floats per x row
#define TILE_ROWS  8              // rows staged per LDS tile
#define CHUNK_ROWS 128            // rows per workgroup
#define N_CHUNKS   8              // 1024 / 128
#define N_TILES    (CHUNK_ROWS / TILE_ROWS)   // 16

// ---------------------------------------------------------------------------
// CDNA5 async global -> LDS copy (GVS mode: SGPR base + 32-bit VGPR offset).
// Tracked by ASYNCcnt; completion enforced with s_wait_asynccnt.
// ---------------------------------------------------------------------------
__device__ __forceinline__ void async_load_b128_to_lds(unsigned lds_byte_off,
                                                       const float* sgpr_base,
                                                       unsigned gbyte_off) {
  asm volatile("global_load_async_to_lds_b128 %0, %1, %2"
               :
               : "v"(lds_byte_off), "v"(gbyte_off), "s"(sgpr_base)
               : "memory");
}

__device__ __forceinline__ void wait_asynccnt_8() {
  asm volatile("s_wait_asynccnt 8" ::: "memory");
}
__device__ __forceinline__ void wait_asynccnt_0() {
  asm volatile("s_wait_asynccnt 0" ::: "memory");
}

// wave32 argmax reduction, first-index tie-break (matches jnp.argmax).
__device__ __forceinline__ void wave32_argmax(float& v, int& idx) {
#pragma unroll
  for (int off = 16; off >= 1; off >>= 1) {
    float ov = __shfl_xor(v, off, 32);
    int   oi = __shfl_xor(idx, off, 32);
    if (ov > v || (ov == v && oi < idx)) { v = ov; idx = oi; }
  }
}

// ---------------------------------------------------------------------------
// Fused single-read pass:
//   grid = 64 matrices * 8 row-chunks = 512 blocks, 256 threads (8 wave32).
//   Double-buffered async staging of 8x1024 f32 tiles into LDS, then:
//     * wave w reduces row w of the tile (row argmax -> written immediately)
//     * thread t accumulates column partials for columns {t, t+256, t+512, t+768}
// ---------------------------------------------------------------------------
__global__ void __launch_bounds__(256) fused_argmax_kernel(
    const float* __restrict__ x,
    int*   __restrict__ row_idx,      // ws: [64*1024]
    float* __restrict__ out_rowidx,   // d_out[0 : 65536)
    float* __restrict__ pv,           // ws: [64][8][1024] column partial maxima
    int*   __restrict__ pi) {         // ws: [64][8][1024] column partial argmax
  __shared__ __align__(16) float tile[2][TILE_ROWS * HW_DIM];   // 2 x 32 KB

  const int t    = threadIdx.x;
  const int lane = t & 31;
  const int wid  = t >> 5;                 // 0..7: row-in-tile owned by this wave
  const int n     = blockIdx.x >> 3;       // matrix id 0..63
  const int chunk = blockIdx.x & 7;        // 128-row chunk 0..7
  const int bh  = n >> 2;
  const int f1i = (n >> 1) & 1;
  const int f2i = n & 1;

  // Byte offset of C[n][chunk*128][0] inside x (fits in 32 bits: |x| = 2^28 B).
  const unsigned base_b =
      ((unsigned)(bh * 2048 + f1i * 1024 + chunk * CHUNK_ROWS) * 2048u +
       (unsigned)(f2i * 1024)) * 4u;
  const unsigned lds0 = (unsigned)(uintptr_t)(&tile[0][0]);

  float cbv[4];
  int   cbi[4];
#pragma unroll
  for (int c = 0; c < 4; ++c) { cbv[c] = -__builtin_inff(); cbi[c] = 0; }

  // Prefetch tile 0 into buffer 0: 8 async b128 ops per wave (one per row),
  // each lane moves 16 bytes; the block covers 8 rows x 4 KB = 32 KB.
#pragma unroll
  for (int rr = 0; rr < TILE_ROWS; ++rr) {
    async_load_b128_to_lds(lds0 + (unsigned)(rr * 4096 + t * 16), x,
                           base_b + (unsigned)rr * ROW_BYTES + (unsigned)t * 16u);
  }

  for (int tl = 0; tl < N_TILES; ++tl) {
    const int buf = tl & 1;
    if (tl + 1 < N_TILES) {
      const int nb = (tl + 1) & 1;
#pragma unroll
      for (int rr = 0; rr < TILE_ROWS; ++rr) {
        async_load_b128_to_lds(
            lds0 + (unsigned)(nb * (TILE_ROWS * HW_DIM * 4) + rr * 4096 + t * 16),
            x,
            base_b + (unsigned)((tl + 1) * TILE_ROWS + rr) * ROW_BYTES +
                (unsigned)t * 16u);
      }
      wait_asynccnt_8();   // 16 in flight; current tile's 8 are done
    } else {
      wait_asynccnt_0();
    }
    __syncthreads();       // all waves' staging visible in LDS

    // ---- Row argmax: wave `wid` reduces tile row `wid` (ds_load_b128) ----
    {
      float best = -__builtin_inff();
      int   bidx = 0;
#pragma unroll
      for (int k = 0; k < 8; ++k) {
        const int j0 = lane * 4 + k * 128;
        const float4 v =
            *reinterpret_cast<const float4*>(&tile[buf][wid * HW_DIM + j0]);
        if (v.x > best) { best = v.x; bidx = j0; }
        if (v.y > best) { best = v.y; bidx = j0 + 1; }
        if (v.z > best) { best = v.z; bidx = j0 + 2; }
        if (v.w > best) { best = v.w; bidx = j0 + 3; }
      }
      wave32_argmax(best, bidx);
      if (lane == 0) {
        const int i = chunk * CHUNK_ROWS + tl * TILE_ROWS + wid;
        const int g = n * HW_DIM + i;
        row_idx[g]    = bidx;
        out_rowidx[g] = (float)bidx;
      }
    }

    // ---- Column partial argmax: thread t owns cols t, t+256, t+512, t+768 ----
#pragma unroll
    for (int rr = 0; rr < TILE_ROWS; ++rr) {
      const int i = chunk * CHUNK_ROWS + tl * TILE_ROWS + rr;
#pragma unroll
      for (int c = 0; c < 4; ++c) {
        const float v = tile[buf][rr * HW_DIM + t + c * 256];
        if (v > cbv[c]) { cbv[c] = v; cbi[c] = i; }   // ascending i -> first idx
      }
    }
    __syncthreads();       // protect this buffer before it is re-staged
  }

#pragma unroll
  for (int c = 0; c < 4; ++c) {
    const int col = t + c * 256;
    pv[(n * N_CHUNKS + chunk) * HW_DIM + col] = cbv[c];
    pi[(n * N_CHUNKS + chunk) * HW_DIM + col] = cbi[c];
  }
}

// Merge the 8 per-chunk column partials -> col_idx[n][j]. Chunks are in
// ascending-row order, so strict-greater preserves first-index semantics.
__global__ void col_argmax_merge_kernel(const float* __restrict__ pv,
                                        const int* __restrict__ pi,
                                        int* __restrict__ col_idx) {
  const int id = blockIdx.x * blockDim.x + threadIdx.x;   // 0..65535
  const int n = id >> 10;
  const int j = id & 1023;
  float best = -__builtin_inff();
  int   bidx = 0;
#pragma unroll
  for (int c = 0; c < N_CHUNKS; ++c) {
    const float v  = pv[(n * N_CHUNKS + c) * HW_DIM + j];
    const int   ix = pi[(n * N_CHUNKS + c) * HW_DIM + j];
    if (v > best) { best = v; bidx = ix; }
  }
  col_idx[id] = bidx;
}

// final_indices[n,i] = col_idx[n, row_idx[n,i]]; mask = (grid distance < 1.5)
// sqrt(d2) < 1.5  <=>  integer d2 <= 2.
__global__ void final_mask_kernel(const int* __restrict__ row_idx,
                                  const int* __restrict__ col_idx,
                                  float* __restrict__ out_mask) {
  const int id = blockIdx.x * blockDim.x + threadIdx.x;   // 0..65535
  const int n = id >> 10;
  const int i = id & 1023;
  const int m = row_idx[id];
  const int j = col_idx[n * HW_DIM + m];
  const int x1 = i >> 5, y1 = i & 31;
  const int x2 = j >> 5, y2 = j & 31;
  const int d2 = (x1 - x2) * (x1 - x2) + (y1 - y2) * (y1 - y2);
  out_mask[id] = (d2 <= 2) ? 1.0f : 0.0f;
}

extern "C" void kernel_launch(void* const* d_in, const int* in_sizes, int n_in,
                              void* d_out, int out_size, void* d_ws, size_t ws_size,
                              hipStream_t stream) {
  const float* x = (const float*)d_in[0];
  float* out = (float*)d_out;            // [0:65536) max_idx, [65536:131072) mask
  char* ws = (char*)d_ws;

  // Workspace layout (4.5 MB total):
  int*   row_idx = (int*)(ws);                              // 256 KB
  int*   col_idx = (int*)(ws + 65536 * 4);                  // 256 KB
  float* pv      = (float*)(ws + 2 * 65536 * 4);            // 2 MB
  int*   pi      = (int*)(ws + 2 * 65536 * 4 + N_MAT * N_CHUNKS * HW_DIM * 4);

  fused_argmax_kernel<<<N_MAT * N_CHUNKS, 256, 0, stream>>>(x, row_idx, out, pv, pi);
  col_argmax_merge_kernel<<<256, 256, 0, stream>>>(pv, pi, col_idx);
  final_mask_kernel<<<256, 256, 0, stream>>>(row_idx, col_idx, out + 65536);
}